// CachedParamMgr_64879775974023
// MI455X (gfx1250) — compile-verified
//
#include <hip/hip_runtime.h>
#include <stdint.h>

// ---------------------------------------------------------------------------
// Reference reduces to: out[i,:] = weight[ids[i],:]  (65536 rows x 128 f32).
// All eviction/scatter bookkeeping cancels out of the returned tensor, and
// inputs must not be mutated, so the kernel is a pure bandwidth-bound gather:
//   ~32 MiB scattered 512B-row reads + ~32 MiB streaming writes + 256 KiB ids
//   => ~2.9 us at 23.3 TB/s HBM. No matmul anywhere => WMMA not applicable.
//
// gfx1250 path: async LDS data mover (ASYNCcnt), VGPR-free staging:
//   global_load_async_to_lds_b128  th:TH_LOAD_NT   (HBM -> LDS)
//   s_wait_asynccnt 0
//   global_store_async_from_lds_b128 th:TH_STORE_NT (LDS -> HBM)
// One wave32 moves one 512B row per b128 instruction (32 lanes x 16B).
//
// IOFFSET trick (CDNA5 async ops add INST_OFFSET to BOTH the LDS and the
// global address): per-row address math is folded into the 24-bit immediate,
// with the gather's global base pre-biased by -r*512 so the shared offset
// lands both sides on the right row. NT hints: rows touched exactly once.
// ---------------------------------------------------------------------------

#define THREADS          256          // 8 waves (wave32)
#define WAVES_PER_BLOCK  8
#define ROWS_PER_WAVE    8
#define ROWS_PER_BLOCK   (WAVES_PER_BLOCK * ROWS_PER_WAVE)   // 64
#define DIM              128
#define ROW_BYTES        (DIM * 4)    // 512

__device__ __forceinline__ void async_gather_row(uint32_t lds_dst, uint64_t gsrc, int imm)
{
    // each lane copies 16 bytes from global address gsrc + imm to LDS address lds_dst + imm
    asm volatile("global_load_async_to_lds_b128 %0, %1, off offset:%c2 th:TH_LOAD_NT"
                 :: "v"(lds_dst), "v"(gsrc), "i"(imm) : "memory");
}

__device__ __forceinline__ void async_store_row(uint64_t gdst, uint32_t lds_src, int imm)
{
    // each lane copies 16 bytes from LDS address lds_src + imm to global address gdst + imm
    asm volatile("global_store_async_from_lds_b128 %0, %1, off offset:%c2 th:TH_STORE_NT"
                 :: "v"(gdst), "v"(lds_src), "i"(imm) : "memory");
}

__device__ __forceinline__ void wait_asynccnt0()
{
    asm volatile("s_wait_asynccnt 0x0" ::: "memory");
}

__global__ __launch_bounds__(THREADS)
void gather_rows_async_kernel(const float* __restrict__ weight,
                              const int*   __restrict__ ids,
                              float*       __restrict__ out,
                              int n_rows)
{
    // 64 rows * 512B = 32 KiB staging (WGP has 320 KiB LDS -> not a limiter).
    __shared__ float smem[ROWS_PER_BLOCK * DIM];

    const int lane    = threadIdx.x & 31;
    const int wave    = threadIdx.x >> 5;
    const int rowBase = blockIdx.x * ROWS_PER_BLOCK + wave * ROWS_PER_WAVE;

    // Low 32 bits of a generic shared pointer == LDS byte offset, which is
    // exactly the VDST/VSRC operand of the async LDS instructions.
    const uint32_t lds0 = (uint32_t)(uintptr_t)(&smem[0])
                        + (uint32_t)(wave * ROWS_PER_WAVE * ROW_BYTES)
                        + (uint32_t)(lane * 16);

    const uint64_t wbase = (uint64_t)(uintptr_t)weight + (uint64_t)(uint32_t)(lane * 16);
    const uint64_t obase = (uint64_t)(uintptr_t)out
                         + (uint64_t)(uint32_t)rowBase * (uint64_t)ROW_BYTES
                         + (uint64_t)(uint32_t)(lane * 16);

    if (rowBase + ROWS_PER_WAVE <= n_rows) {
        // ---------------- fast path: no EXEC juggling --------------------
        // Preload all 8 ids (32B contiguous, rowBase is 8-aligned) as two
        // int4 loads -> both issued before any use, single loadcnt wait.
        const int4* idp = (const int4*)(ids + rowBase);
        const int4 ia = idp[0];
        const int4 ib = idp[1];
        const int idv[ROWS_PER_WAVE] = { ia.x, ia.y, ia.z, ia.w, ib.x, ib.y, ib.z, ib.w };

        // Phase 1: 8 back-to-back async gathers, HBM -> LDS.
        // Global base biased by -r*512 (sign-extended 64-bit); the shared
        // IOFFSET (+r*512) restores global addr AND steps the LDS dest.
        #pragma unroll
        for (int r = 0; r < ROWS_PER_WAVE; ++r) {
            const uint64_t src = wbase
                + (uint64_t)((int64_t)(idv[r] - r) * (int64_t)ROW_BYTES);
            async_gather_row(lds0, src, r * ROW_BYTES);
        }

        wait_asynccnt0();   // all 8 rows landed in LDS

        // Phase 2: 8 back-to-back async stores, LDS -> out (coalesced).
        // One base each side; IOFFSET steps both in lockstep.
        #pragma unroll
        for (int r = 0; r < ROWS_PER_WAVE; ++r) {
            async_store_row(obase, lds0, r * ROW_BYTES);
        }
    } else {
        // ---------------- ragged tail (not taken for 65536 rows) ---------
        #pragma unroll
        for (int r = 0; r < ROWS_PER_WAVE; ++r) {
            const int row = rowBase + r;          // wave-uniform guard
            if (row < n_rows) {
                const int id = ids[row];
                const uint64_t src = wbase + (uint64_t)(uint32_t)id * (uint64_t)ROW_BYTES;
                async_gather_row(lds0 + (uint32_t)(r * ROW_BYTES), src, 0);
            }
        }
        wait_asynccnt0();
        #pragma unroll
        for (int r = 0; r < ROWS_PER_WAVE; ++r) {
            const int row = rowBase + r;
            if (row < n_rows) {
                async_store_row(obase + (uint64_t)(uint32_t)(r * ROW_BYTES),
                                lds0 + (uint32_t)(r * ROW_BYTES), 0);
            }
        }
    }
    // S_ENDPGM performs an implicit wait-idle: ASYNCcnt drains before exit.
}

extern "C" void kernel_launch(void* const* d_in, const int* in_sizes, int n_in,
                              void* d_out, int out_size, void* d_ws, size_t ws_size,
                              hipStream_t stream)
{
    (void)n_in; (void)out_size; (void)d_ws; (void)ws_size;

    // setup_inputs() order:
    //   0: weight               (1,000,000 x 128) f32
    //   1: cuda_cached_weight   (262144 x 128)    f32   (unused: cancels out)
    //   2: cached_idx_map       (262144,)         i32   (unused: cancels out)
    //   3: inverted_cached_idx  (1,000,000,)      i32   (unused: cancels out)
    //   4: ids                  (65536,)          i32
    const float* weight = (const float*)d_in[0];
    const int*   ids    = (const int*)d_in[4];
    float*       out    = (float*)d_out;

    const int n_rows = in_sizes[4];                                      // 65536
    const int grid   = (n_rows + ROWS_PER_BLOCK - 1) / ROWS_PER_BLOCK;   // 1024

    gather_rows_async_kernel<<<grid, THREADS, 0, stream>>>(weight, ids, out, n_rows);
}